// Model_5463198400662
// MI455X (gfx1250) — compile-verified
//
#include <hip/hip_runtime.h>

#define NN 100000
#define EE 1600000
#define DD 128

typedef __attribute__((ext_vector_type(2))) float v2f;
typedef __attribute__((ext_vector_type(8))) float v8f;

// ---------------- degree / norm ----------------

__global__ void k_init_deg(float* deg) {
    int i = blockIdx.x * blockDim.x + threadIdx.x;
    if (i < NN) deg[i] = 1.0f;   // self-loop weight
}

__global__ void k_deg_scatter(const int* __restrict__ ei,
                              const float* __restrict__ ew,
                              float* __restrict__ deg) {
    int e = blockIdx.x * blockDim.x + threadIdx.x;
    if (e < EE) {
        int c = ei[EE + e];              // col
        unsafeAtomicAdd(&deg[c], ew[e]); // global_atomic_add_f32
    }
}

__global__ void k_dinv(float* deg) {
    int i = blockIdx.x * blockDim.x + threadIdx.x;
    if (i < NN) {
        float d = deg[i];
        deg[i] = (d > 0.0f) ? rsqrtf(d) : 0.0f;
    }
}

// ---------------- WMMA fp32 GEMM: O[100000x128] = A[100000x128] * W[128x128] ----
// grid = 6250 blocks (16 rows each), block = 256 threads = 8 waves,
// wave w computes the 16x16 tile at columns [16w, 16w+16).

__global__ void k_gemm_wmma(const float* __restrict__ A,
                            const float* __restrict__ W,
                            float* __restrict__ O) {
    const int lane  = threadIdx.x & 31;
    const int wave  = threadIdx.x >> 5;    // 0..7 -> N tile
    const int half  = lane >> 4;           // 0 or 1
    const int l     = lane & 15;
    const int mbase = blockIdx.x * 16;
    const int nbase = wave * 16;

    v8f acc = {};
    const float* arow = A + (size_t)(mbase + l) * DD;

#pragma unroll
    for (int k = 0; k < DD; k += 4) {
        const int kk = k + half * 2;       // this half-wave covers K = kk, kk+1
        // A fragment (16x4): lane l, half h -> A[mbase+l][kk], A[mbase+l][kk+1]
        v2f a;
        a.x = arow[kk];
        a.y = arow[kk + 1];
        // B fragment (4x16): lane l, half h -> W[kk][nbase+l], W[kk+1][nbase+l]
        v2f b;
        b.x = W[(size_t)kk * DD + nbase + l];
        b.y = W[(size_t)(kk + 1) * DD + nbase + l];
        acc = __builtin_amdgcn_wmma_f32_16x16x4_f32(
            false, a, false, b, (short)0, acc, false, false);
    }

    // C/D layout: VGPR r, lanes 0-15 -> M=r, lanes 16-31 -> M=r+8; N = l
    float* orow = O + (size_t)(mbase + half * 8) * DD + nbase + l;
#pragma unroll
    for (int r = 0; r < 8; ++r)
        orow[(size_t)r * DD] = acc[r];
}

// ---------------- accumulator init / scatter / finalize ----------------

// acc = h*dinv^2 + b   (self-loop contribution + bias folded in)
__global__ void k_init_acc(const float* __restrict__ h,
                           const float* __restrict__ dinv,
                           const float* __restrict__ bias,
                           float* __restrict__ acc) {
    long t = (long)blockIdx.x * blockDim.x + threadIdx.x;
    if (t < (long)NN * DD) {
        int i = (int)(t >> 7);
        int d = (int)(t & 127);
        float s = dinv[i];
        acc[t] = h[t] * s * s + bias[d];
    }
}

// out = h*dinv^2 + b + x   (layer-2 init, bias + residual folded in)
__global__ void k_init_out(const float* __restrict__ h,
                           const float* __restrict__ dinv,
                           const float* __restrict__ bias,
                           const float* __restrict__ x,
                           float* __restrict__ out) {
    long t = (long)blockIdx.x * blockDim.x + threadIdx.x;
    if (t < (long)NN * DD) {
        int i = (int)(t >> 7);
        int d = (int)(t & 127);
        float s = dinv[i];
        out[t] = h[t] * s * s + bias[d] + x[t];
    }
}

// One wave32 per edge: gather 512B row of src, atomically add norm-scaled
// message into acc[col]. Edge metadata is wave-uniform -> scalar loads.
__global__ void k_edge_scatter(const int* __restrict__ ei,
                               const float* __restrict__ ew,
                               const float* __restrict__ dinv,
                               const float* __restrict__ src,
                               float* __restrict__ acc) {
    int edge = blockIdx.x * 8 + (threadIdx.x >> 5);
    if (edge >= EE) return;
    int lane = threadIdx.x & 31;

    int r = ei[edge];        // row (source node)
    int c = ei[EE + edge];   // col (destination node)
    float norm = dinv[r] * ew[edge] * dinv[c];

    const float4* s = (const float4*)(src + (size_t)r * DD);
    float4 v = s[lane];
    float* d = acc + (size_t)c * DD + lane * 4;
    unsafeAtomicAdd(d + 0, v.x * norm);
    unsafeAtomicAdd(d + 1, v.y * norm);
    unsafeAtomicAdd(d + 2, v.z * norm);
    unsafeAtomicAdd(d + 3, v.w * norm);
}

__global__ void k_relu(const float* __restrict__ acc, float* __restrict__ h) {
    long t = (long)blockIdx.x * blockDim.x + threadIdx.x;
    if (t < (long)NN * DD) {
        float v = acc[t];
        h[t] = v > 0.0f ? v : 0.0f;
    }
}

// ---------------- launch ----------------

extern "C" void kernel_launch(void* const* d_in, const int* in_sizes, int n_in,
                              void* d_out, int out_size, void* d_ws, size_t ws_size,
                              hipStream_t stream) {
    const float* x  = (const float*)d_in[0];   // [N,D]
    const float* w1 = (const float*)d_in[1];   // [D,D]
    const float* b1 = (const float*)d_in[2];   // [D]
    const float* w2 = (const float*)d_in[3];   // [D,D]
    const float* b2 = (const float*)d_in[4];   // [D]
    const float* ew = (const float*)d_in[5];   // [E]
    const int*   ei = (const int*)d_in[6];     // [2,E]
    float* out = (float*)d_out;                // [N,D]

    float* dinv = (float*)d_ws;                // N floats (deg -> dinv in place)
    float* P1   = dinv + NN;                   // N*D floats: h1, later relu(h)
    float* P2   = P1 + (size_t)NN * DD;        // N*D floats: accumulator / h@w2

    const int T = 256;
    const int gN  = (NN + T - 1) / T;          // 391
    const int gE  = EE / T;                    // 6250
    const int gND = (NN * DD) / T;             // 50000
    const int gM  = NN / 16;                   // 6250 (exact)
    const int gS  = EE / 8;                    // 200000 (8 edges/block)

    // normalization
    k_init_deg<<<gN, T, 0, stream>>>(dinv);
    k_deg_scatter<<<gE, T, 0, stream>>>(ei, ew, dinv);
    k_dinv<<<gN, T, 0, stream>>>(dinv);

    // layer 1: h1 = x@w1 ; acc = self + b1 ; scatter ; h = relu(acc)
    k_gemm_wmma<<<gM, T, 0, stream>>>(x, w1, P1);
    k_init_acc<<<gND, T, 0, stream>>>(P1, dinv, b1, P2);
    k_edge_scatter<<<gS, T, 0, stream>>>(ei, ew, dinv, P1, P2);
    k_relu<<<gND, T, 0, stream>>>(P2, P1);

    // layer 2: h2 = h@w2 ; out = self + b2 + x ; scatter
    k_gemm_wmma<<<gM, T, 0, stream>>>(P1, w2, P2);
    k_init_out<<<gND, T, 0, stream>>>(P2, dinv, b2, x, out);
    k_edge_scatter<<<gS, T, 0, stream>>>(ei, ew, dinv, P2, out);
}